// Catting_75058848465342
// MI455X (gfx1250) — compile-verified
//
#include <hip/hip_runtime.h>

// Transpose the last two axes of x[16, 64, 64, 512] (f32):
//   out[s, i*64 + j] = x[s, j, i],   s = b*64+c in [0,1024), j in [0,64), i in [0,512)
//
// Pure bandwidth problem (268 MB traffic ~= 11.5 us at 23.3 TB/s).
// Strategy: 64x64 tile per 256-thread block, filled with CDNA5 async
// global->LDS B128 loads (ASYNCcnt), XOR-swizzled LDS layout for
// conflict-free transposed reads, coalesced B128 stores.

#define N_COLS   512   // dim 3 of x (i)
#define S_ROWS   64    // dim 2 of x (j)
#define TILE     64
#define LDS_STRIDE 68  // dwords per tile row: 272 B (16B-aligned chunks, bank-rotating)

__global__ __launch_bounds__(256) void Catting_75058848465342_kernel(
    const float* __restrict__ in, float* __restrict__ out) {
  __shared__ float tile[TILE * LDS_STRIDE];  // 17,408 bytes

  const unsigned t     = threadIdx.x;   // 0..255
  const unsigned slice = blockIdx.x;    // 0..1023  (= b*64 + c)
  const unsigned itile = blockIdx.y;    // 0..7     (i-tile index)

  const float* __restrict__ src = in  + (size_t)slice * (S_ROWS * N_COLS) + itile * TILE;
  float*       __restrict__ dst = out + (size_t)slice * (S_ROWS * N_COLS) + itile * (TILE * S_ROWS);

  // ---- Fill phase: async global -> LDS, 16B per lane per iteration ----
  // lane t, iter: row j = iter*16 + t/16, chunk c = t%16 (i = 4c).
  // Global side: 16 lanes cover one contiguous 256B row segment.
  // LDS side: chunk placed at j*68 + 4*(c ^ ((j>>2)&15))  (XOR swizzle).
  {
    const unsigned c  = t & 15u;
    const unsigned jh = t >> 4;   // 0..15
#pragma unroll
    for (unsigned iter = 0; iter < 4; ++iter) {
      const unsigned j = iter * 16u + jh;
      const float* gp = src + (size_t)j * N_COLS + 4u * c;
      const unsigned ldsdw = j * LDS_STRIDE + 4u * (c ^ ((j >> 2) & 15u));
      const unsigned ldsb  = (unsigned)(unsigned long long)(&tile[ldsdw]);
      asm volatile("global_load_async_to_lds_b128 %0, %1, off"
                   :
                   : "v"(ldsb), "v"(gp)
                   : "memory");
    }
  }

  // Each wave drains its own async queue, then the workgroup barrier makes
  // every wave's LDS writes visible to all.
  asm volatile("s_wait_asynccnt 0x0" ::: "memory");
  __syncthreads();

  // ---- Drain phase: transposed scalar LDS reads -> coalesced B128 stores ----
  // lane t, iter: output row i = iter*16 + t/16, j4 = 4*(t%16).
  // Reads tile[(j4+k)][i], k=0..3; with the swizzle, (j4+k)>>2 == t%16 for all k,
  // so the column offset is constant per lane and banks are conflict-free.
  {
    const unsigned m  = t & 15u;          // j4 = 4m
    const unsigned ih = t >> 4;           // 0..15
#pragma unroll
    for (unsigned iter = 0; iter < 4; ++iter) {
      const unsigned i   = iter * 16u + ih;
      const unsigned col = 4u * ((i >> 2) ^ m) + (i & 3u);
      const unsigned r0  = (4u * m + 0u) * LDS_STRIDE + col;
      float4 v;
      v.x = tile[r0];
      v.y = tile[r0 + LDS_STRIDE];
      v.z = tile[r0 + 2 * LDS_STRIDE];
      v.w = tile[r0 + 3 * LDS_STRIDE];
      *reinterpret_cast<float4*>(dst + (size_t)i * TILE + 4u * m) = v;
    }
  }
}

extern "C" void kernel_launch(void* const* d_in, const int* in_sizes, int n_in,
                              void* d_out, int out_size, void* d_ws, size_t ws_size,
                              hipStream_t stream) {
  (void)in_sizes; (void)n_in; (void)out_size; (void)d_ws; (void)ws_size;
  const float* x = (const float*)d_in[0];
  float* out = (float*)d_out;
  // 16*64 = 1024 (B,C) slices, 512/64 = 8 i-tiles each.
  dim3 grid(1024, 8, 1);
  dim3 block(256, 1, 1);
  Catting_75058848465342_kernel<<<grid, block, 0, stream>>>(x, out);
}